// DeNetBoth_31963146617489
// MI455X (gfx1250) — compile-verified
//
#include <hip/hip_runtime.h>
#include <hip/hip_bf16.h>

typedef __attribute__((ext_vector_type(2))) float v2f;
typedef __attribute__((ext_vector_type(8))) float v8f;
typedef __attribute__((ext_vector_type(4))) unsigned int v4u;
typedef __attribute__((ext_vector_type(4))) int v4i;
typedef __attribute__((ext_vector_type(8))) int v8i;

#define HH 128
#define WWI 128
#define HW (HH * WWI)

// ---------------------------------------------------------------------------
// TDM support detection. ROCm 7.2 clang-22: 5-arg builtin; amdgpu-toolchain
// (therock-10.0 headers present): 6-arg builtin.
// ---------------------------------------------------------------------------
#if defined(__has_builtin)
#if __has_builtin(__builtin_amdgcn_tensor_load_to_lds)
#define HAVE_TDM 1
#endif
#endif
#if defined(__has_include)
#if __has_include(<hip/amd_detail/amd_gfx1250_TDM.h>)
#define TDM_ARITY6 1
#endif
#endif

__device__ __forceinline__ v8f wmma_f32_4(v2f a, v2f b, v8f c) {
  // D = A(16x4 f32) * B(4x16 f32) + C(16x16 f32)
  return __builtin_amdgcn_wmma_f32_16x16x4_f32(
      /*neg_a=*/false, a, /*neg_b=*/false, b,
      /*c_mod=*/(short)0, c, /*reuse_a=*/false, /*reuse_b=*/false);
}

#if defined(HAVE_TDM)
// Issue a TDM load of a 2D tile of 4-byte elements into LDS.
//   tileW x tileH elements, source rows strideW elements apart.
// D# packing per CDNA5 ISA 8.3/8.4 (group0: count/lds/global/type,
// group1: data_size, tensor dims, tile dims, dim0 stride). Groups 2/3 zero
// (tensor <= 2D). Executed once per wave (TDM ignores EXEC).
__device__ __forceinline__ void tdm_load_to_lds(unsigned lds_off, const float* gptr,
                                                unsigned tileW, unsigned tileH,
                                                unsigned tensorW, unsigned strideW) {
  unsigned long long ga = (unsigned long long)(uintptr_t)gptr;
  v4u g0;
  g0[0] = 1u;                                    // count=1, user-mode, no gather
  g0[1] = lds_off;                               // lds_addr (bytes)
  g0[2] = (unsigned)ga;                          // global_addr[31:0]
  g0[3] = (unsigned)(ga >> 32) | (2u << 30);     // global_addr[56:32] | type=2
  v8i g1;
  g1[0] = (int)(2u << 16);                       // data_size = 2 (4 bytes)
  g1[1] = (int)((tensorW & 0xffffu) << 16);      // tensor_dim0[15:0]
  g1[2] = (int)((tensorW >> 16) | ((tileH & 0xffffu) << 16));  // dim0[31:16]|dim1[15:0]
  g1[3] = (int)((tileH >> 16) | (tileW << 16));  // dim1[31:16] | tile_dim0
  g1[4] = (int)(tileH & 0xffffu);                // tile_dim1 (tile_dim2 = 0)
  g1[5] = (int)strideW;                          // tensor_dim0_stride[31:0]
  g1[6] = 0;
  g1[7] = 0;
  v4i z4;
  z4[0] = z4[1] = z4[2] = z4[3] = 0;
#if defined(TDM_ARITY6)
  v8i z8;
  z8[0] = z8[1] = z8[2] = z8[3] = z8[4] = z8[5] = z8[6] = z8[7] = 0;
  __builtin_amdgcn_tensor_load_to_lds(g0, g1, z4, z4, z8, 0);
#else
  __builtin_amdgcn_tensor_load_to_lds(g0, g1, z4, z4, 0);
#endif
}
#endif  // HAVE_TDM

// ---------------------------------------------------------------------------
// One-shot weight reshapes (device side, tiny) into TDM-friendly layouts.
// ---------------------------------------------------------------------------
// wT[tap][co][ci], co padded to CP with zeros: contiguous 2D tiles per tap.
__global__ __launch_bounds__(256) void transpose_conv_w(
    const float* __restrict__ w, float* __restrict__ wT,
    int CIN, int COUT, int CP) {
  int idx = blockIdx.x * 256 + threadIdx.x;
  int total = 9 * CP * CIN;
  if (idx >= total) return;
  int tap = idx / (CP * CIN);
  int rem = idx % (CP * CIN);
  int co = rem / CIN;
  int ci = rem % CIN;
  wT[idx] = (co < COUT) ? w[(co * CIN + ci) * 9 + tap] : 0.f;
}

// wdT[o][tap][c] (o = global out channel): each group's panel is contiguous.
__global__ __launch_bounds__(256) void transpose_dcn_w(
    const float* __restrict__ wd, float* __restrict__ wdT) {
  int idx = blockIdx.x * 256 + threadIdx.x;
  if (idx >= 96 * 288) return;
  int o = idx / 288;
  int rem = idx % 288;
  int tap = rem >> 5;
  int c = rem & 31;
  wdT[idx] = wd[(o * 32 + c) * 9 + tap];
}

// ---------------------------------------------------------------------------
// Generic 3x3 conv (stride 1, pad 1) as implicit GEMM with fp32 WMMA.
//   M = COUT (tiles of 16), N = 16 consecutive pixels per wave, K = CIN*9
//   K ordering: k = tap*CIN + ci  (chunks of 4 never straddle a tap)
// Weights arrive pre-transposed/padded: wT[tap][CP][CIN]; per-(tap,cbase)
// slices are DMA'd into LDS by the Tensor Data Mover.
// CONCAT=true: input channels are cat(in0[0:96], in1[0:96]) (bottom conv).
// ---------------------------------------------------------------------------
template <int CIN, int COUT, bool CONCAT>
__global__ __launch_bounds__(128) void conv3x3_wmma_kernel(
    const float* __restrict__ in0, const float* __restrict__ in1,
    const float* __restrict__ wT,  // [9, CP, CIN] zero-padded
    float* __restrict__ out)       // [COUT, H, W]
{
  constexpr int MT = (COUT + 15) / 16;  // M tiles
  constexpr int CP = MT * 16;           // padded Cout
  constexpr int CSTEP = (CIN > 96) ? 96 : CIN;
  __shared__ float ldsw[CP * CSTEP];    // per-(tap, cbase) weight slice

  const int tid = threadIdx.x;
  const int wave = tid >> 5;
  const int lane = tid & 31;
  const int half = lane >> 4;
  const int lm = lane & 15;

  const int tile = blockIdx.x * 4 + wave;  // 1024 tiles of 16 pixels
  const int y = tile >> 3;
  const int x0 = (tile & 7) << 4;
  const int px = x0 + lm;

  v8f acc[MT];
#pragma unroll
  for (int mt = 0; mt < MT; ++mt)
#pragma unroll
    for (int r = 0; r < 8; ++r) acc[mt][r] = 0.f;

  for (int tap = 0; tap < 9; ++tap) {
    const int dy = tap / 3 - 1;
    const int dx = tap % 3 - 1;
    const int yy = y + dy;
    const int xx = px + dx;
    const bool vy = (unsigned)yy < (unsigned)HH;
    const bool vx = (unsigned)xx < (unsigned)WWI;
    const int yc = vy ? yy : 0;
    const int xc = vx ? xx : 0;
    const float bscale = (vy && vx) ? 1.f : 0.f;
    const int sp = yc * WWI + xc;

    for (int cbase = 0; cbase < CIN; cbase += CSTEP) {
      __syncthreads();  // previous slice fully consumed
#if defined(HAVE_TDM)
      if (tid < 32) {  // wave 0 issues the DMA and waits on TENSORcnt
        tdm_load_to_lds((unsigned)(uintptr_t)&ldsw[0],
                        wT + (size_t)(tap * CP) * CIN + cbase,
                        /*tileW=*/CSTEP, /*tileH=*/CP,
                        /*tensorW=*/CIN, /*strideW=*/CIN);
        __builtin_amdgcn_s_wait_tensorcnt(0);
      }
#else
      for (int idx = tid; idx < CP * CSTEP; idx += 128) {
        const int co = idx / CSTEP;
        const int j = idx % CSTEP;
        ldsw[idx] = wT[(size_t)(tap * CP + co) * CIN + cbase + j];
      }
#endif
      __syncthreads();  // slice visible to all waves

      for (int c0 = 0; c0 < CSTEP; c0 += 4) {
        const int cc = cbase + c0 + half * 2;  // this lane's K channels (cc, cc+1)
        float bx, by;
        if (CONCAT) {
          const float* s0 = (cc < 96) ? (in0 + cc * HW) : (in1 + (cc - 96) * HW);
          const float* s1 = (cc + 1 < 96) ? (in0 + (cc + 1) * HW) : (in1 + (cc + 1 - 96) * HW);
          bx = s0[sp] * bscale;
          by = s1[sp] * bscale;
        } else {
          bx = in0[cc * HW + sp] * bscale;
          by = in0[(cc + 1) * HW + sp] * bscale;
        }
        v2f b;
        b.x = bx;
        b.y = by;
#pragma unroll
        for (int mt = 0; mt < MT; ++mt) {
          v2f a = *(const v2f*)&ldsw[(mt * 16 + lm) * CSTEP + (c0 + half * 2)];
          acc[mt] = wmma_f32_4(a, b, acc[mt]);
        }
      }
    }
  }

  // D layout: lane, vgpr r -> M = half*8 + r, N = lm
  const int pidx = y * WWI + x0 + lm;
#pragma unroll
  for (int mt = 0; mt < MT; ++mt) {
#pragma unroll
    for (int r = 0; r < 8; ++r) {
      const int co = mt * 16 + half * 8 + r;
      if (co < COUT) out[co * HW + pidx] = acc[mt][r];
    }
  }
}

// ---------------------------------------------------------------------------
// Modulated deformable conv (DCNv2), one group per blockIdx.y.
// Fuses: sigmoid(mask), bilinear sampling, modulation, grouped 3x3 conv as
// GEMM (M=32 out-ch -> two 16x16 WMMA accumulators, K = tap*32 + c), bias.
// Group weight panel (36 KB, contiguous in wdT) is DMA'd to LDS by the TDM.
// ---------------------------------------------------------------------------
__global__ __launch_bounds__(128) void dcn_wmma_kernel(
    const float* __restrict__ feat,    // [96, H, W]
    const float* __restrict__ offraw,  // [81, H, W]
    const float* __restrict__ wdT,     // [96, 9, 32]  (= [o][tap][c])
    const float* __restrict__ bd,      // [96]
    float* __restrict__ out)           // [96, H, W]
{
  __shared__ float ldsw[32 * 288];  // [o][tap*32 + c]
  const int g = blockIdx.y;
  const int tid = threadIdx.x;
  const int wave = tid >> 5;
  const int lane = tid & 31;
  const int half = lane >> 4;
  const int lm = lane & 15;

#if defined(HAVE_TDM)
  if (tid < 32) {
    tdm_load_to_lds((unsigned)(uintptr_t)&ldsw[0], wdT + (size_t)g * 32 * 288,
                    /*tileW=*/32 * 288, /*tileH=*/1,
                    /*tensorW=*/32 * 288, /*strideW=*/32 * 288);
    __builtin_amdgcn_s_wait_tensorcnt(0);
  }
#else
  for (int idx = tid; idx < 32 * 288; idx += 128)
    ldsw[idx] = wdT[(size_t)g * 32 * 288 + idx];
#endif
  __syncthreads();

  const int tile = blockIdx.x * 4 + wave;
  const int y = tile >> 3;
  const int x0 = (tile & 7) << 4;
  const int p = x0 + lm;
  const int pidx = y * WWI + p;

  v8f acc0, acc1;
#pragma unroll
  for (int r = 0; r < 8; ++r) {
    acc0[r] = 0.f;
    acc1[r] = 0.f;
  }

  const float* fg = feat + g * 32 * HW;

  for (int tap = 0; tap < 9; ++tap) {
    const int och = g * 9 + tap;
    const float dyv = offraw[och * HW + pidx];
    const float dxv = offraw[(27 + och) * HW + pidx];
    const float mraw = offraw[(54 + och) * HW + pidx];
    const float mv = 1.f / (1.f + __expf(-mraw));

    const float py = (float)y + (float)(tap / 3 - 1) + dyv;
    const float pxs = (float)p + (float)(tap % 3 - 1) + dxv;
    const float fy0 = floorf(py);
    const float fx0 = floorf(pxs);
    const int iy0 = (int)fy0;
    const int ix0 = (int)fx0;
    const float fy = py - fy0;
    const float fx = pxs - fx0;

    const bool vy0 = (unsigned)iy0 < (unsigned)HH;
    const bool vy1 = (unsigned)(iy0 + 1) < (unsigned)HH;
    const bool vx0 = (unsigned)ix0 < (unsigned)WWI;
    const bool vx1 = (unsigned)(ix0 + 1) < (unsigned)WWI;
    const int cy0 = vy0 ? iy0 : 0;
    const int cy1 = vy1 ? (iy0 + 1) : 0;
    const int cx0 = vx0 ? ix0 : 0;
    const int cx1 = vx1 ? (ix0 + 1) : 0;

    const float w00 = (1.f - fy) * (1.f - fx) * ((vy0 && vx0) ? mv : 0.f);
    const float w01 = (1.f - fy) * fx * ((vy0 && vx1) ? mv : 0.f);
    const float w10 = fy * (1.f - fx) * ((vy1 && vx0) ? mv : 0.f);
    const float w11 = fy * fx * ((vy1 && vx1) ? mv : 0.f);
    const int i00 = cy0 * WWI + cx0;
    const int i01 = cy0 * WWI + cx1;
    const int i10 = cy1 * WWI + cx0;
    const int i11 = cy1 * WWI + cx1;

    for (int c0 = 0; c0 < 32; c0 += 4) {
      const int cA = c0 + half * 2;  // this lane's two K channels
      const float* f0 = fg + cA * HW;
      const float* f1 = f0 + HW;
      v2f b;
      b.x = w00 * f0[i00] + w01 * f0[i01] + w10 * f0[i10] + w11 * f0[i11];
      b.y = w00 * f1[i00] + w01 * f1[i01] + w10 * f1[i10] + w11 * f1[i11];
      v2f a0 = *(const v2f*)&ldsw[(0 * 16 + lm) * 288 + tap * 32 + cA];
      v2f a1 = *(const v2f*)&ldsw[(16 + lm) * 288 + tap * 32 + cA];
      acc0 = wmma_f32_4(a0, b, acc0);
      acc1 = wmma_f32_4(a1, b, acc1);
    }
  }

#pragma unroll
  for (int r = 0; r < 8; ++r) {
    const int m = half * 8 + r;
    const int co0 = g * 32 + m;
    const int co1 = g * 32 + 16 + m;
    out[co0 * HW + pidx] = acc0[r] + bd[co0];
    out[co1 * HW + pidx] = acc1[r] + bd[co1];
  }
}

// ---------------------------------------------------------------------------
// Only batch element n=1 of the reference's 4 stacked frames reaches the
// output (all ops are batch-independent), so we compute just that image:
// feat = conv3x3(cat(X1[0], X2[0]), w_bot), then 4x (offset conv -> DCN).
// ---------------------------------------------------------------------------
extern "C" void kernel_launch(void* const* d_in, const int* in_sizes, int n_in,
                              void* d_out, int out_size, void* d_ws, size_t ws_size,
                              hipStream_t stream) {
  const float* X1 = (const float*)d_in[0];  // [2,96,128,128]; use batch 0
  const float* X2 = (const float*)d_in[1];
  const float* w_bot = (const float*)d_in[2];
  const float* w_off[4] = {(const float*)d_in[3], (const float*)d_in[6],
                           (const float*)d_in[9], (const float*)d_in[12]};
  const float* w_d[4] = {(const float*)d_in[4], (const float*)d_in[7],
                         (const float*)d_in[10], (const float*)d_in[13]};
  const float* b_d[4] = {(const float*)d_in[5], (const float*)d_in[8],
                         (const float*)d_in[11], (const float*)d_in[14]};

  float* featA = (float*)d_ws;
  float* featB = featA + 96 * HW;
  float* offraw = featB + 96 * HW;
  float* wT_bot = offraw + 81 * HW;          // 9*96*192
  float* wT_off = wT_bot + 9 * 96 * 192;     // 9*96*96
  float* wdT = wT_off + 9 * 96 * 96;         // 96*288

  dim3 blk(128);

  // Bottom conv: Cin=192 (cat), Cout=96.
  transpose_conv_w<<<(9 * 96 * 192 + 255) / 256, 256, 0, stream>>>(w_bot, wT_bot, 192, 96, 96);
  conv3x3_wmma_kernel<192, 96, true><<<256, blk, 0, stream>>>(X1, X2, wT_bot, featA);

  float* cur = featA;
  float* nxt = featB;
  for (int i = 0; i < 4; ++i) {
    // Offset conv: Cin=96, Cout=81 (raw; sigmoid fused into DCN).
    transpose_conv_w<<<(9 * 96 * 96 + 255) / 256, 256, 0, stream>>>(w_off[i], wT_off, 96, 81, 96);
    conv3x3_wmma_kernel<96, 81, false><<<256, blk, 0, stream>>>(cur, nullptr, wT_off, offraw);
    transpose_dcn_w<<<(96 * 288 + 255) / 256, 256, 0, stream>>>(w_d[i], wdT);
    float* dst = (i == 3) ? (float*)d_out : nxt;
    dcn_wmma_kernel<<<dim3(256, 3), blk, 0, stream>>>(cur, offraw, wdT, b_d[i], dst);
    float* t = cur;
    cur = nxt;
    nxt = t;
  }
}